// ScaleGNN_84129819394303
// MI455X (gfx1250) — compile-verified
//
#include <hip/hip_runtime.h>
#include <math.h>

// ---------------------------------------------------------------------------
// ScaleGNN 2-layer GCN forward for MI455X (gfx1250, wave32, WMMA).
// Dense GEMMs use V_WMMA_F32_16X16X4_F32 (fp32 matrix path, 16x16 tiles).
// W is staged in LDS pre-paired (float2 of K,K+1 rows) so each B fragment is
// a single aligned ds_load_b64 into the WMMA operand pair (no repack movs).
// Edge aggregation is L2-resident f32 atomic scatter (h fits in 192MB L2).
// ---------------------------------------------------------------------------

typedef float v2f __attribute__((ext_vector_type(2)));
typedef float v8f __attribute__((ext_vector_type(8)));

#define HDIM 64
#define THR_LCS 0.1f
#define LN_EPS 1e-5f

static inline int ceil_div(int a, int b) { return (a + b - 1) / b; }

// ---------------------------------------------------------------- init ------
__global__ void k_init(float* indeg, unsigned* mm, int np) {
  int t = blockIdx.x * blockDim.x + threadIdx.x;
  if (t < np) indeg[t] = 0.0f;
  if (t == 0) { mm[0] = 0x7F800000u; /* +inf (min slot) */ mm[1] = 0u; /* max */ }
}

// ------------------------------------------------------------- degree -------
__global__ void k_degree(const int* __restrict__ dst, float* indeg, int E) {
  int e = blockIdx.x * blockDim.x + threadIdx.x;
  if (e < E) atomicAdd(&indeg[dst[e]], 1.0f);
}

// ------------------------------------------------- WMMA fp32 GEMM -----------
// Hout[nrows x 64] = X[nrows x K] @ W[K x 64]
// One wave per 16-row strip; 4 column tiles of 16.
// LDS holds W as float2 pairs: sW[p*64+n] = {W[2p][n], W[2p+1][n]}.
__global__ void k_gemm_wmma_f32(const float* __restrict__ X,
                                const float* __restrict__ W,
                                float* __restrict__ Hout,
                                int nrows, int K, int nStrips) {
  __shared__ v2f sW[64 * HDIM];               // (K/2)*64 float2, max 32 KB
  const int tid = threadIdx.x;
  // cooperative staging: element i = W[k][n] -> sW[(k>>1)*64+n][k&1]
  for (int i = tid; i < K * HDIM; i += blockDim.x) {
    int k = i >> 6, n = i & 63;
    ((float*)sW)[(((k >> 1) * HDIM + n) << 1) + (k & 1)] = W[i];
  }
  __syncthreads();

  const int strip = blockIdx.x * (blockDim.x >> 5) + (tid >> 5); // wave-uniform
  if (strip < nStrips) {                       // uniform branch: EXEC stays all-1
    const int lane = tid & 31;
    const int m    = lane & 15;
    const int kb   = (lane >> 4) << 1;         // 0 or 2 (A/B K split per ISA)
    int row  = strip * 16 + m;
    int rowc = row < nrows ? row : nrows - 1;  // clamp loads; discard at store
    const float* xr = X + (size_t)rowc * K;

    v8f c0 = {0,0,0,0,0,0,0,0}, c1 = c0, c2 = c0, c3 = c0;

    for (int k0 = 0; k0 < K; k0 += 4) {
      v2f a;
      a.x = xr[k0 + kb];                       // merged global_load_b64
      a.y = xr[k0 + kb + 1];
      // B fragment: one ds_load_b64 per tile, imm offsets 0/128/256/384 B
      const v2f* wp = sW + (((k0 + kb) >> 1) * HDIM + m);
      v2f b0 = wp[ 0];
      v2f b1 = wp[16];
      v2f b2 = wp[32];
      v2f b3 = wp[48];
      c0 = __builtin_amdgcn_wmma_f32_16x16x4_f32(false, a, false, b0, (short)0, c0, false, false);
      c1 = __builtin_amdgcn_wmma_f32_16x16x4_f32(false, a, false, b1, (short)0, c1, false, false);
      c2 = __builtin_amdgcn_wmma_f32_16x16x4_f32(false, a, false, b2, (short)0, c2, false, false);
      c3 = __builtin_amdgcn_wmma_f32_16x16x4_f32(false, a, false, b3, (short)0, c3, false, false);
    }

    // C/D layout: VGPR r -> lanes 0-15: M=r, lanes 16-31: M=r+8; N = lane&15
    const int nIdx  = lane & 15;
    const int mBase = strip * 16 + ((lane >> 4) ? 8 : 0);
#pragma unroll
    for (int r = 0; r < 8; ++r) {
      int orow = mBase + r;
      if (orow < nrows) {
        float* o = Hout + (size_t)orow * HDIM + nIdx;
        o[ 0] = c0[r]; o[16] = c1[r]; o[32] = c2[r]; o[48] = c3[r];
      }
    }
  }
}

// ----------------------------------------- self-loop + bias (layer 1) -------
// agg[n,c] = h[n,c] / (indeg[n]+1) + b[c]   (initializes the scatter target)
__global__ void k_self1(const float* __restrict__ h, const float* __restrict__ indeg,
                        const float* __restrict__ b, float* __restrict__ agg, int N) {
  int t = blockIdx.x * blockDim.x + threadIdx.x;
  if (t >= N * HDIM) return;
  int n = t >> 6, c = t & 63;
  agg[t] = h[t] / (indeg[n] + 1.0f) + b[c];
}

// --------------------------------------------- edge scatter (layer 1) -------
// thread = (edge, 4 channels); float4 gather + 4 f32 atomics (L2-resident)
__global__ void k_agg1(const int* __restrict__ src, const int* __restrict__ dst,
                       const float* __restrict__ h, const float* __restrict__ indeg,
                       float* __restrict__ agg, int E) {
  int t = blockIdx.x * blockDim.x + threadIdx.x;
  if (t >= E * 16) return;
  int e = t >> 4, c4 = (t & 15) << 2;
  int s = src[e], d = dst[e];
  float w = rsqrtf(indeg[s] + 1.0f) * rsqrtf(indeg[d] + 1.0f);
  float4 hv = *reinterpret_cast<const float4*>(h + (size_t)s * HDIM + c4);
  float* o = agg + (size_t)d * HDIM + c4;
  atomicAdd(o + 0, hv.x * w);
  atomicAdd(o + 1, hv.y * w);
  atomicAdd(o + 2, hv.z * w);
  atomicAdd(o + 3, hv.w * w);
}

// --------------------------- fused LayerNorm + ReLU + L2 score (wave/node) --
__global__ void k_ln_relu_score(float* __restrict__ hbuf,
                                const float* __restrict__ gamma,
                                const float* __restrict__ beta,
                                float* __restrict__ sArr, unsigned* mm, int N) {
  __shared__ unsigned bMin, bMax;
  if (threadIdx.x == 0) { bMin = 0x7F800000u; bMax = 0u; }
  __syncthreads();

  int wid  = (blockIdx.x * blockDim.x + threadIdx.x) >> 5;   // node id
  int lane = threadIdx.x & 31;
  if (wid < N) {
    float* row = hbuf + (size_t)wid * HDIM;
    float v0 = row[lane], v1 = row[lane + 32];
    float sum = v0 + v1;
#pragma unroll
    for (int o = 16; o > 0; o >>= 1) sum += __shfl_xor(sum, o, 32);
    float mu = sum * (1.0f / 64.0f);
    float d0 = v0 - mu, d1 = v1 - mu;
    float sq = d0 * d0 + d1 * d1;
#pragma unroll
    for (int o = 16; o > 0; o >>= 1) sq += __shfl_xor(sq, o, 32);
    float rstd = rsqrtf(sq * (1.0f / 64.0f) + LN_EPS);
    float h0 = fmaxf(d0 * rstd * gamma[lane]      + beta[lane],      0.0f);
    float h1 = fmaxf(d1 * rstd * gamma[lane + 32] + beta[lane + 32], 0.0f);
    row[lane] = h0; row[lane + 32] = h1;
    float nsq = h0 * h0 + h1 * h1;
#pragma unroll
    for (int o = 16; o > 0; o >>= 1) nsq += __shfl_xor(nsq, o, 32);
    if (lane == 0) {
      float sv = sqrtf(nsq);
      sArr[wid] = sv;
      unsigned b = __float_as_uint(sv);      // sv >= 0: uint order == float order
      atomicMin(&bMin, b);
      atomicMax(&bMax, b);
    }
  }
  __syncthreads();
  if (threadIdx.x == 0) { atomicMin(&mm[0], bMin); atomicMax(&mm[1], bMax); }
}

// -------------------------- self-loop + bias (layer 2, masked degree) -------
// deg2[n] = 1 + mask[n]*indeg[n]  (mask depends only on dst node!)
__global__ void k_self2(const float* __restrict__ h2, const float* __restrict__ indeg,
                        const float* __restrict__ sArr, const unsigned* __restrict__ mm,
                        const float* __restrict__ b, float* __restrict__ out, int N) {
  int t = blockIdx.x * blockDim.x + threadIdx.x;
  if (t >= N * HDIM) return;
  int n = t >> 6, c = t & 63;
  float smin = __uint_as_float(mm[0]), smax = __uint_as_float(mm[1]);
  float den  = smax - smin + 1e-8f;
  float mk   = ((sArr[n] - smin) / den > THR_LCS) ? 1.0f : 0.0f;
  float deg2 = 1.0f + mk * indeg[n];
  out[t] = h2[t] / deg2 + b[c];
}

// ------------------------------------- masked edge scatter (layer 2) --------
__global__ void k_agg2(const int* __restrict__ src, const int* __restrict__ dst,
                       const float* __restrict__ h2, const float* __restrict__ indeg,
                       const float* __restrict__ sArr, const unsigned* __restrict__ mm,
                       float* __restrict__ out, int E) {
  int t = blockIdx.x * blockDim.x + threadIdx.x;
  if (t >= E * 16) return;
  int e = t >> 4, c4 = (t & 15) << 2;
  int s = src[e], d = dst[e];
  float smin = __uint_as_float(mm[0]), smax = __uint_as_float(mm[1]);
  float den  = smax - smin + 1e-8f;
  if ((sArr[d] - smin) / den <= THR_LCS) return;      // ew = 0: no contribution
  float deg2d = 1.0f + indeg[d];                      // mask[d] == 1 here
  float mks   = ((sArr[s] - smin) / den > THR_LCS) ? 1.0f : 0.0f;
  float deg2s = 1.0f + mks * indeg[s];
  float w = rsqrtf(deg2s) * rsqrtf(deg2d);
  float4 hv = *reinterpret_cast<const float4*>(h2 + (size_t)s * HDIM + c4);
  float* o = out + (size_t)d * HDIM + c4;
  atomicAdd(o + 0, hv.x * w);
  atomicAdd(o + 1, hv.y * w);
  atomicAdd(o + 2, hv.z * w);
  atomicAdd(o + 3, hv.w * w);
}

// --------------------------------------- in-place log-softmax (wave/node) ---
__global__ void k_log_softmax(float* __restrict__ out, int N) {
  int wid  = (blockIdx.x * blockDim.x + threadIdx.x) >> 5;
  int lane = threadIdx.x & 31;
  if (wid >= N) return;
  float* row = out + (size_t)wid * HDIM;
  float v0 = row[lane], v1 = row[lane + 32];
  float m = fmaxf(v0, v1);
#pragma unroll
  for (int o = 16; o > 0; o >>= 1) m = fmaxf(m, __shfl_xor(m, o, 32));
  float s = __expf(v0 - m) + __expf(v1 - m);
#pragma unroll
  for (int o = 16; o > 0; o >>= 1) s += __shfl_xor(s, o, 32);
  float lse = m + __logf(s);
  row[lane] = v0 - lse;
  row[lane + 32] = v1 - lse;
}

// ---------------------------------------------------------------------------
extern "C" void kernel_launch(void* const* d_in, const int* in_sizes, int n_in,
                              void* d_out, int out_size, void* d_ws, size_t ws_size,
                              hipStream_t stream) {
  const float* x     = (const float*)d_in[0];
  const float* W1    = (const float*)d_in[1];
  const float* b1    = (const float*)d_in[2];
  const float* W2    = (const float*)d_in[3];
  const float* b2    = (const float*)d_in[4];
  const float* gamma = (const float*)d_in[5];
  const float* beta  = (const float*)d_in[6];
  const int*   ei    = (const int*)d_in[7];

  const int IN = 128;
  const int N  = in_sizes[0] / IN;       // 100000
  const int E  = in_sizes[7] / 2;        // 1000000
  const int* src = ei;
  const int* dst = ei + E;

  // ---- workspace layout (all offsets 256B aligned) ----
  const size_t NP = (size_t)((N + 255) & ~255);
  float*    indeg = (float*)d_ws;                       // NP floats
  float*    sArr  = indeg + NP;                         // NP floats
  unsigned* mm    = (unsigned*)(sArr + NP);             // 2 used, 64 reserved
  float*    bufA  = (float*)(mm + 64);                  // NP*64 (h1 / h2)
  float*    bufB  = bufA + NP * HDIM;                   // NP*64 (agg1 / h_ln)
  float*    out   = (float*)d_out;
  (void)ws_size; (void)n_in; (void)out_size;

  const int B = 256;
  const int nStrips = ceil_div(N, 16);
  const int gemmGrid = ceil_div(nStrips, B / 32);

  // 1. init degree + min/max slots
  k_init<<<ceil_div((int)NP, B), B, 0, stream>>>(indeg, mm, (int)NP);
  // 2. in-degree
  k_degree<<<ceil_div(E, B), B, 0, stream>>>(dst, indeg, E);
  // 3. h1 = x @ W1  (WMMA fp32)
  k_gemm_wmma_f32<<<gemmGrid, B, 0, stream>>>(x, W1, bufA, N, IN, nStrips);
  // 4. agg1 = h1/deg + b1 (self loop)
  k_self1<<<ceil_div(N * HDIM, B), B, 0, stream>>>(bufA, indeg, b1, bufB, N);
  // 5. agg1 += scatter(h1[src] * norm)
  k_agg1<<<ceil_div(E * 16, B), B, 0, stream>>>(src, dst, bufA, indeg, bufB, E);
  // 6. LayerNorm + ReLU + node score (in place on bufB)
  k_ln_relu_score<<<ceil_div(N * 32, B), B, 0, stream>>>(bufB, gamma, beta, sArr, mm, N);
  // 7. h2 = h_ln @ W2  (WMMA fp32)
  k_gemm_wmma_f32<<<gemmGrid, B, 0, stream>>>(bufB, W2, bufA, N, HDIM, nStrips);
  // 8. out = h2/deg2 + b2 (masked self loop; deg2 = 1 + mask*indeg)
  k_self2<<<ceil_div(N * HDIM, B), B, 0, stream>>>(bufA, indeg, sArr, mm, b2, out, N);
  // 9. out += masked scatter
  k_agg2<<<ceil_div(E * 16, B), B, 0, stream>>>(src, dst, bufA, indeg, sArr, mm, out, E);
  // 10. log_softmax in place
  k_log_softmax<<<ceil_div(N * 32, B), B, 0, stream>>>(out, N);
}